// GCNLayer_90486370992279
// MI455X (gfx1250) — compile-verified
//
#include <hip/hip_runtime.h>
#include <hip/hip_bf16.h>

typedef __attribute__((ext_vector_type(2))) float v2f;
typedef __attribute__((ext_vector_type(8))) float v8f;

#define D_FEAT 128

// ---------------------------------------------------------------------------
// Kernel 1: zero the accumulation buffer (d_out doubles as h).
// ---------------------------------------------------------------------------
__global__ void gcn_zero_kernel(float4* __restrict__ p, int n4) {
    int i = blockIdx.x * blockDim.x + threadIdx.x;
    if (i < n4) p[i] = make_float4(0.f, 0.f, 0.f, 0.f);
}

// ---------------------------------------------------------------------------
// Kernel 2: edge scatter-add. One wave32 per edge; each lane moves 4 floats.
// x rows (25.6MB) live in L2; atomics are non-returning f32 adds to L2.
// ---------------------------------------------------------------------------
__global__ void gcn_scatter_kernel(const float* __restrict__ x,
                                   const int* __restrict__ src,
                                   const int* __restrict__ dst,
                                   float* __restrict__ h,
                                   int n_edges) {
    int wave_in_block = threadIdx.x >> 5;
    int lane = threadIdx.x & 31;
    int e = blockIdx.x * (blockDim.x >> 5) + wave_in_block;
    if (e >= n_edges) return;

    int s = src[e];   // uniform per wave -> scalar load
    int d = dst[e];

    const float4* xs = (const float4*)(x + (size_t)s * D_FEAT);
    float4 v = xs[lane];                       // 32 lanes x 16B = 512B row

    float* hd = h + (size_t)d * D_FEAT + lane * 4;
    atomicAdd(hd + 0, v.x);
    atomicAdd(hd + 1, v.y);
    atomicAdd(hd + 2, v.z);
    atomicAdd(hd + 3, v.w);
}

// ---------------------------------------------------------------------------
// Kernel 3: in-place linear layer out = h @ W^T + b using fp32 WMMA.
// Block = 256 threads (8 waves). Block b owns rows [16b, 16b+16):
//   stage 16x128 h-stripe to LDS -> barrier -> 8 waves each compute a 16x16
//   tile (obase = wave*16) with 32x V_WMMA_F32_16X16X4_F32 -> overwrite rows.
// ---------------------------------------------------------------------------
__global__ void gcn_linear_wmma_kernel(float* __restrict__ h_out,
                                       const float* __restrict__ Wm,
                                       const float* __restrict__ bias) {
    __shared__ float sh[16 * D_FEAT];          // 8 KB

    const int tid  = threadIdx.x;
    const int row0 = blockIdx.x * 16;

    // Stage this block's 16x128 fp32 stripe into LDS (512 float4, 2/thread).
    const float4* src4 = (const float4*)(h_out + (size_t)row0 * D_FEAT);
    float4* s4 = (float4*)sh;
    s4[tid]       = src4[tid];
    s4[tid + 256] = src4[tid + 256];
    __syncthreads();

    const int wave  = tid >> 5;
    const int lane  = tid & 31;
    const int t     = lane & 15;   // M index for A, N index for B/C/D
    const int g     = lane >> 4;   // lane-group: K offset (A/B), row offset (C/D)
    const int obase = wave * 16;   // this wave's output-column tile

    // C init: bias broadcast along rows (C[m][n] = bias[obase+n]).
    float bval = bias[obase + t];
    v8f c = { bval, bval, bval, bval, bval, bval, bval, bval };

    // A lane base: h row t, k = 2g + {0,1} per VGPR pair.
    const float* arow = sh + t * D_FEAT + 2 * g;
    // B lane base: B[k][n] = W[obase+n][k] -> row (obase+t) of W, same k offsets.
    const float* wrow = Wm + (size_t)(obase + t) * D_FEAT + 2 * g;

#pragma unroll
    for (int kb = 0; kb < D_FEAT / 4; ++kb) {
        v2f a = *(const v2f*)(arow + kb * 4);
        v2f b = *(const v2f*)(wrow + kb * 4);
        // 8 args: (neg_a, A, neg_b, B, c_mod, C, reuse_a, reuse_b)
        c = __builtin_amdgcn_wmma_f32_16x16x4_f32(
                false, a, false, b, (short)0, c, false, false);
    }

    // D layout: VGPR j, group g -> row M = j + 8g, col N = t.
#pragma unroll
    for (int j = 0; j < 8; ++j) {
        h_out[(size_t)(row0 + j + 8 * g) * D_FEAT + obase + t] = c[j];
    }
}

// ---------------------------------------------------------------------------
extern "C" void kernel_launch(void* const* d_in, const int* in_sizes, int n_in,
                              void* d_out, int out_size, void* d_ws, size_t ws_size,
                              hipStream_t stream) {
    const float* x   = (const float*)d_in[0];
    const int*   src = (const int*)d_in[1];
    const int*   dst = (const int*)d_in[2];
    const float* Wm  = (const float*)d_in[3];
    const float* b   = (const float*)d_in[4];
    float* out = (float*)d_out;

    const int n_nodes = in_sizes[0] / D_FEAT;   // 50000
    const int n_edges = in_sizes[1];            // 800000

    // 1) zero accumulator (in d_out)
    int n4 = n_nodes * (D_FEAT / 4);
    gcn_zero_kernel<<<(n4 + 255) / 256, 256, 0, stream>>>((float4*)out, n4);

    // 2) scatter-add edge messages: one wave per edge, 8 edges per block
    int edge_blocks = (n_edges + 7) / 8;
    gcn_scatter_kernel<<<edge_blocks, 256, 0, stream>>>(x, src, dst, out, n_edges);

    // 3) in-place fp32 WMMA linear: one block per 16-row stripe
    gcn_linear_wmma_kernel<<<n_nodes / 16, 256, 0, stream>>>(out, Wm, b);
}